// HistoricCurrent_72026601554051
// MI455X (gfx1250) — compile-verified
//
#include <hip/hip_runtime.h>

typedef float v2f __attribute__((ext_vector_type(2)));
typedef float v8f __attribute__((ext_vector_type(8)));

namespace {

constexpr int kB  = 256;   // batch
constexpr int kS  = 512;   // sequence
constexpr int kE  = 512;   // embed
constexpr int kH  = 512;   // hidden
constexpr int kG  = 2048;  // 4*H gates
constexpr int kTF = 799;
constexpr int kTFP = 800;  // padded K for tf-idf GEMM
constexpr int kSub = 512;
constexpr int kH1 = 256;
constexpr int kH2 = 128;

constexpr int BLK_M = 32;
constexpr int BLK_N = 128;
constexpr int KC    = 32;   // K panel
constexpr int AP    = 36;   // LDS stride for A panel [m][k] (pad, 16B-alignable)
constexpr int BPT   = 34;   // LDS stride for transposed B panel [n][k] (even pad)
constexpr int NT    = 256;  // 8 waves (wave32)

__device__ __forceinline__ float sigf(float x) {
  return 1.f / (1.f + __expf(-x));
}

// ---- wave-level WMMA inner product over one (A,B) segment -----------------
// A panel staged [m][k] (stride AP); B panel staged transposed [n][k]
// (stride BPT) so both fragments are contiguous float2 -> ds_load_b64.
__device__ __forceinline__ void mm_panel(const float* __restrict__ A, long lda,
                                         const float* __restrict__ Bm, long ldb,
                                         int m0, int n0, int K,
                                         float* As, float* BsT,
                                         v8f& acc0, v8f& acc1)
{
  const int tid   = threadIdx.x;
  const int lane  = tid & 31;
  const int wave  = tid >> 5;
  const int waveM = wave >> 2;                 // 0..1
  const int waveN = wave & 3;                  // 0..3
  const int ml    = (waveM << 4) + (lane & 15);// A-fragment row within block
  const int kb    = (lane >> 4) << 1;          // 0 (lanes 0-15) / 2 (lanes 16-31)
  const int nl    = (waveN << 5) + (lane & 15);// B/C column within block

  const int arow = tid >> 3;                   // 0..31
  const int akk  = (tid & 7) << 2;             // 0,4,...,28
  const int bkr  = tid >> 5;                   // 0..7
  const int bnc  = (tid & 31) << 2;            // 0..124

  const float* aptr  = As  + ml * AP + kb;
  const float* bptr0 = BsT + nl * BPT + kb;
  const float* bptr1 = BsT + (nl + 16) * BPT + kb;

  for (int k0 = 0; k0 < K; k0 += KC) {
    __syncthreads();
    {   // stage A: 32 rows x 32 k, coalesced float4
      const float4 v = *(const float4*)(A + (long)(m0 + arow) * lda + (k0 + akk));
      float* dst = As + arow * AP + akk;
      dst[0] = v.x; dst[1] = v.y; dst[2] = v.z; dst[3] = v.w;
    }
    #pragma unroll
    for (int r = 0; r < 4; ++r) {   // stage B: 32 k x 128 n, transposed into LDS
      const int krow = bkr + r * 8;
      const float4 v = *(const float4*)(Bm + (long)(k0 + krow) * ldb + (n0 + bnc));
      BsT[(bnc + 0) * BPT + krow] = v.x;
      BsT[(bnc + 1) * BPT + krow] = v.y;
      BsT[(bnc + 2) * BPT + krow] = v.z;
      BsT[(bnc + 3) * BPT + krow] = v.w;
    }
    __syncthreads();
    #pragma unroll
    for (int kk = 0; kk < KC; kk += 4) {
      const v2f a  = *(const v2f*)(aptr  + kk);
      const v2f b0 = *(const v2f*)(bptr0 + kk);
      const v2f b1 = *(const v2f*)(bptr1 + kk);
      acc0 = __builtin_amdgcn_wmma_f32_16x16x4_f32(false, a, false, b0,
                                                   (short)0, acc0, false, false);
      acc1 = __builtin_amdgcn_wmma_f32_16x16x4_f32(false, a, false, b1,
                                                   (short)0, acc1, false, false);
    }
  }
}

// ---- write C fragments with bias + activation -----------------------------
// act: 0 = none, 1 = relu, 2 = sigmoid, 3 = tanh
__device__ __forceinline__ void store_tile(float* __restrict__ Cm, long ldc,
                                           int m0, int n0,
                                           const float* __restrict__ bias0,
                                           const float* __restrict__ bias1,
                                           int act, v8f acc0, v8f acc1)
{
  const int tid   = threadIdx.x;
  const int lane  = tid & 31;
  const int wave  = tid >> 5;
  const int waveM = wave >> 2;
  const int waveN = wave & 3;
  const int col   = n0 + (waveN << 5) + (lane & 15);
  const int rowb  = m0 + (waveM << 4) + ((lane >> 4) << 3);
  const float b00 = bias0 ? bias0[col]      : 0.f;
  const float b01 = bias1 ? bias1[col]      : 0.f;
  const float b10 = bias0 ? bias0[col + 16] : 0.f;
  const float b11 = bias1 ? bias1[col + 16] : 0.f;
  #pragma unroll
  for (int r = 0; r < 8; ++r) {
    float v0 = acc0[r] + b00 + b01;
    float v1 = acc1[r] + b10 + b11;
    if (act == 1)      { v0 = fmaxf(v0, 0.f); v1 = fmaxf(v1, 0.f); }
    else if (act == 2) { v0 = sigf(v0);       v1 = sigf(v1); }
    else if (act == 3) { v0 = tanhf(v0);      v1 = tanhf(v1); }
    Cm[(long)(rowb + r) * ldc + col]      = v0;
    Cm[(long)(rowb + r) * ldc + col + 16] = v1;
  }
}

} // namespace

// ---- one LSTM time-step, split-K across blocks ----------------------------
// bj 0..15 : gates0 = h   @ W_all + b_W   (raw partial)
// bj 16..31: gates1 = x_t @ U_all + b_U   (raw partial)
// bj 32..35: cs1    = c   @ W_d   + b_d   (raw, tanh applied in update)
// Splitting h/x segments across blocks halves the dependent-WMMA chain per
// step (128 deep instead of 256) and widens the per-step grid to 288 blocks.
__global__ void __launch_bounds__(NT)
lstm_step_gemm(const float* __restrict__ h, const float* __restrict__ x,
               const float* __restrict__ c,
               const float* __restrict__ Wall, const float* __restrict__ Uall,
               const float* __restrict__ Wd,
               const float* __restrict__ Wall_b, const float* __restrict__ Uall_b,
               const float* __restrict__ Wd_b,
               float* __restrict__ gates0, float* __restrict__ gates1,
               float* __restrict__ cs1, int t)
{
  __shared__ float As[BLK_M * AP];
  __shared__ float BsT[BLK_N * BPT];
  const int bm = blockIdx.x & 7;        // 256/32 = 8 row tiles
  const int bj = blockIdx.x >> 3;       // 0..35 job id
  const int m0 = bm * BLK_M;
  v8f acc0 = {0.f, 0.f, 0.f, 0.f, 0.f, 0.f, 0.f, 0.f};
  v8f acc1 = acc0;
  if (bj < 16) {
    const int n0 = bj * BLK_N;
    mm_panel(h, kH, Wall, kG, m0, n0, kH, As, BsT, acc0, acc1);
    store_tile(gates0, kG, m0, n0, Wall_b, nullptr, 0, acc0, acc1);
  } else if (bj < 32) {
    const int n0 = (bj - 16) * BLK_N;
    mm_panel(x + (long)t * kE, (long)kS * kE, Uall, kG, m0, n0, kE, As, BsT, acc0, acc1);
    store_tile(gates1, kG, m0, n0, Uall_b, nullptr, 0, acc0, acc1);
  } else {
    const int n0 = (bj - 32) * BLK_N;
    mm_panel(c, kH, Wd, kH, m0, n0, kH, As, BsT, acc0, acc1);
    store_tile(cs1, kH, m0, n0, Wd_b, nullptr, 0, acc0, acc1);
  }
}

// ---- generic GEMM with optional second K segment (for concat inputs) ------
__global__ void __launch_bounds__(NT)
gemm_bias_act(const float* __restrict__ A0, long lda0, int K0,
              const float* __restrict__ Bm0,
              const float* __restrict__ A1, long lda1, int K1,
              const float* __restrict__ Bm1,
              const float* __restrict__ bias0, const float* __restrict__ bias1,
              float* __restrict__ Cm, int N, int act)
{
  __shared__ float As[BLK_M * AP];
  __shared__ float BsT[BLK_N * BPT];
  const int m0 = blockIdx.x * BLK_M;
  const int n0 = blockIdx.y * BLK_N;
  v8f acc0 = {0.f, 0.f, 0.f, 0.f, 0.f, 0.f, 0.f, 0.f};
  v8f acc1 = acc0;
  mm_panel(A0, lda0, Bm0, N, m0, n0, K0, As, BsT, acc0, acc1);
  if (A1) mm_panel(A1, lda1, Bm1, N, m0, n0, K1, As, BsT, acc0, acc1);
  store_tile(Cm, N, m0, n0, bias0, bias1, act, acc0, acc1);
}

// ---- elementwise LSTM state update (activations fused here) ---------------
__global__ void lstm_update(const float* __restrict__ g0,
                            const float* __restrict__ g1,
                            const float* __restrict__ dpre,
                            const float* __restrict__ ts, int t,
                            float* __restrict__ h, float* __restrict__ c,
                            float* __restrict__ hsum)
{
  const int idx = blockIdx.x * blockDim.x + threadIdx.x;   // B*H
  const int bb = idx >> 9;
  const int j  = idx & 511;
  const long o = (long)bb * kG;
  const float f  = sigf(g0[o + j]           + g1[o + j]);
  const float i  = sigf(g0[o + kH + j]      + g1[o + kH + j]);
  const float oo = sigf(g0[o + 2 * kH + j]  + g1[o + 2 * kH + j]);
  const float ct = sigf(g0[o + 3 * kH + j]  + g1[o + 3 * kH + j]);
  const float c1 = tanhf(dpre[idx]);
  const float tt = ts[bb * kS + t];
  const float cold = c[idx];
  const float cadj = (cold - c1) + c1 * tt;
  const float cn = f * cadj + i * ct;
  const float hn = oo * tanhf(cn);
  c[idx] = cn;
  h[idx] = hn;
  hsum[idx] += hn;
}

// ---- small helpers --------------------------------------------------------
__global__ void fill_zero(float* __restrict__ p, int n) {
  const int i = blockIdx.x * blockDim.x + threadIdx.x;
  if (i < n) p[i] = 0.f;
}
__global__ void scale_copy(const float* __restrict__ s, float* __restrict__ d,
                           float f, int n) {
  const int i = blockIdx.x * blockDim.x + threadIdx.x;
  if (i < n) d[i] = s[i] * f;
}
__global__ void pad_tfidf(const float* __restrict__ src, float* __restrict__ dst) {
  const int i = blockIdx.x * blockDim.x + threadIdx.x;
  if (i >= kB * kTFP) return;
  const int b = i / kTFP, k = i % kTFP;
  dst[i] = (k < kTF) ? src[b * kTF + k] : 0.f;
}
__global__ void pad_fcw(const float* __restrict__ src, float* __restrict__ dst) {
  const int i = blockIdx.x * blockDim.x + threadIdx.x;
  if (i >= kTFP * kSub) return;
  const int k = i / kSub, n = i % kSub;
  dst[i] = (k < kTF) ? src[k * kSub + n] : 0.f;
}
__global__ void final_out(const float* __restrict__ a2,
                          const float* __restrict__ w,
                          const float* __restrict__ b,
                          float* __restrict__ out) {
  const int idx = blockIdx.x * blockDim.x + threadIdx.x;
  if (idx >= kB * 2) return;
  const int bb = idx >> 1, j = idx & 1;
  float s = b[j];
  for (int k = 0; k < kH2; ++k) s += a2[bb * kH2 + k] * w[k * 2 + j];
  out[idx] = s;
}

extern "C" void kernel_launch(void* const* d_in, const int* in_sizes, int n_in,
                              void* d_out, int out_size, void* d_ws, size_t ws_size,
                              hipStream_t stream)
{
  const float* x      = (const float*)d_in[0];
  const float* ts     = (const float*)d_in[1];
  const float* tfidf  = (const float*)d_in[2];
  const float* Wall_w = (const float*)d_in[3];
  const float* Wall_b = (const float*)d_in[4];
  const float* Uall_w = (const float*)d_in[5];
  const float* Uall_b = (const float*)d_in[6];
  const float* Wd_w   = (const float*)d_in[7];
  const float* Wd_b   = (const float*)d_in[8];
  const float* fcw    = (const float*)d_in[9];
  const float* fcb    = (const float*)d_in[10];
  const float* l1w    = (const float*)d_in[11];
  const float* l1b    = (const float*)d_in[12];
  const float* l2w    = (const float*)d_in[13];
  const float* l2b    = (const float*)d_in[14];
  const float* low    = (const float*)d_in[15];
  const float* lob    = (const float*)d_in[16];
  float* out = (float*)d_out;

  const int BH = kB * kH; // 131072
  float* ws     = (float*)d_ws;
  float* h      = ws;                        // BH
  float* c      = h      + BH;               // BH
  float* hsum   = c      + BH;               // BH
  float* hmean  = hsum   + BH;               // BH
  float* gates0 = hmean  + BH;               // B*4H
  float* gates1 = gates0 + (long)kB * kG;    // B*4H
  float* cs1    = gates1 + (long)kB * kG;    // BH
  float* sub    = cs1    + BH;               // BH
  float* a1     = sub    + BH;               // B*256
  float* a2     = a1     + kB * kH1;         // B*128
  float* tpad   = a2     + kB * kH2;         // B*800
  float* wpad   = tpad   + kB * kTFP;        // 800*512
  // total ~10 MB of f32 scratch (all step state stays L2-resident)

  // state init (harness does not re-zero between replays)
  fill_zero<<<(3 * BH) / 256, 256, 0, stream>>>(h, 3 * BH);

  // TF-IDF branch (independent of the recurrence)
  pad_tfidf<<<(kB * kTFP + 255) / 256, 256, 0, stream>>>(tfidf, tpad);
  pad_fcw<<<(kTFP * kSub + 255) / 256, 256, 0, stream>>>(fcw, wpad);
  gemm_bias_act<<<dim3(kB / BLK_M, kSub / BLK_N), NT, 0, stream>>>(
      tpad, kTFP, kTFP, wpad, nullptr, 0, 0, nullptr,
      fcb, nullptr, sub, kSub, /*relu*/1);

  // time-LSTM recurrence: 512 sequential steps, 2 launches each
  for (int t = 0; t < kS; ++t) {
    lstm_step_gemm<<<288, NT, 0, stream>>>(h, x, c, Wall_w, Uall_w, Wd_w,
                                           Wall_b, Uall_b, Wd_b,
                                           gates0, gates1, cs1, t);
    lstm_update<<<BH / 256, 256, 0, stream>>>(gates0, gates1, cs1, ts, t,
                                              h, c, hsum);
  }

  // mean over time, then MLP head
  scale_copy<<<BH / 256, 256, 0, stream>>>(hsum, hmean, 1.f / (float)kS, BH);
  gemm_bias_act<<<dim3(kB / BLK_M, kH1 / BLK_N), NT, 0, stream>>>(
      sub, kSub, kSub, l1w,
      hmean, kH, kH, l1w + (long)kSub * kH1,
      l1b, nullptr, a1, kH1, /*relu*/1);
  gemm_bias_act<<<dim3(kB / BLK_M, kH2 / BLK_N), NT, 0, stream>>>(
      a1, kH1, kH1, l2w, nullptr, 0, 0, nullptr,
      l2b, nullptr, a2, kH2, /*relu*/1);
  final_out<<<2, 256, 0, stream>>>(a2, low, lob, out);
}